// DCFConsciousnessDynamics_55886114455830
// MI455X (gfx1250) — compile-verified
//
#include <hip/hip_runtime.h>
#include <stdint.h>

#define BLOCK   1024          // 32 waves of wave32 -> fills a WGP
#define VMAX    50257         // vocab size; one full row lives in LDS (201 KB < 320 KB)
#define NSTEPS  10
#define KUR_DT  0.1f

#if __has_builtin(__builtin_amdgcn_global_load_async_to_lds_b128) && \
    __has_builtin(__builtin_amdgcn_global_load_async_to_lds_b32)
#define USE_ASYNC_BUILTIN 1
#else
#define USE_ASYNC_BUILTIN 0
#endif

// Exact pointee types per hipcc diagnostics:
//   b128: param0 = '__vector_size__(4*sizeof(int)) int __device__ *'
//   b32 : param0 = 'int __device__ *'
typedef int v4i_t __attribute__((__vector_size__(4 * sizeof(int))));
typedef __attribute__((address_space(1))) v4i_t* gptr_v4i;
typedef __attribute__((address_space(3))) v4i_t* lptr_v4i;
typedef __attribute__((address_space(1))) int*   gptr_i32;
typedef __attribute__((address_space(3))) int*   lptr_i32;

__device__ __forceinline__ void async_wait_zero() {
#if __has_builtin(__builtin_amdgcn_s_wait_asynccnt)
  __builtin_amdgcn_s_wait_asynccnt(0);
#else
  asm volatile("s_wait_asynccnt 0" ::: "memory");
#endif
}

// Two-value block reduction for a 1024-thread (32-wave, wave32) block.
// Wave-level xor-shuffle tree, then wave0 reduces the 32 partials.
__device__ __forceinline__ void blockReduce2(float& a, float& b,
                                             float* redA, float* redB,
                                             float* bcast) {
  const int lane = threadIdx.x & 31;
  const int wave = threadIdx.x >> 5;
#pragma unroll
  for (int off = 16; off; off >>= 1) {
    a += __shfl_xor(a, off, 32);
    b += __shfl_xor(b, off, 32);
  }
  __syncthreads();  // protect scratch from previous reduction round
  if (lane == 0) { redA[wave] = a; redB[wave] = b; }
  __syncthreads();
  if (wave == 0) {
    a = redA[lane];
    b = redB[lane];
#pragma unroll
    for (int off = 16; off; off >>= 1) {
      a += __shfl_xor(a, off, 32);
      b += __shfl_xor(b, off, 32);
    }
    if (lane == 0) { bcast[0] = a; bcast[1] = b; }
  }
  __syncthreads();
  a = bcast[0];
  b = bcast[1];
}

__global__ __launch_bounds__(BLOCK)
void kuramoto_rowresident(const float* __restrict__ logits,
                          const float* __restrict__ omega,
                          const float* __restrict__ noise,
                          float* __restrict__ out,
                          int V) {
  // Entire row of phases is LDS-resident for all 10 steps + epilogue.
  __shared__ alignas(16) float ph[VMAX];
  __shared__ float redA[32];
  __shared__ float redB[32];
  __shared__ float bcast[2];

  const int    tid  = threadIdx.x;
  const size_t base = (size_t)blockIdx.x * (size_t)V;
  const float* __restrict__ gno  = noise  + base;
  const float* __restrict__ glg  = logits + base;
  float*       __restrict__ gout = out    + base;

  const int nchunk = V >> 2;          // # of 16-byte chunks (12564)
  const int vtail  = nchunk << 2;     // first tail element (50256)

  // ---- Stage phases_noise row into LDS via async global->LDS copies ----
  {
#if USE_ASYNC_BUILTIN
    __attribute__((address_space(1))) char* g1 =
        (__attribute__((address_space(1))) char*)gno;
    __attribute__((address_space(3))) char* l3 =
        (__attribute__((address_space(3))) char*)ph;
    for (int ci = tid; ci < nchunk; ci += BLOCK) {
      __builtin_amdgcn_global_load_async_to_lds_b128(
          (gptr_v4i)(g1 + 16u * (unsigned)ci),
          (lptr_v4i)(l3 + 16u * (unsigned)ci),
          0, 0);
      __builtin_prefetch(glg + 4 * ci, 0, 3);  // warm logits (global_prefetch_b8)
    }
    for (int i = vtail + tid; i < V; i += BLOCK) {
      __builtin_amdgcn_global_load_async_to_lds_b32(
          (gptr_i32)(g1 + 4u * (unsigned)i),
          (lptr_i32)(l3 + 4u * (unsigned)i),
          0, 0);
    }
#else
    unsigned lbase =
        (unsigned)(uintptr_t)(__attribute__((address_space(3))) char*)ph;
    for (int ci = tid; ci < nchunk; ci += BLOCK) {
      asm volatile("global_load_async_to_lds_b128 %0, %1, off"
                   :
                   : "v"(lbase + 16u * (unsigned)ci), "v"(gno + 4 * ci)
                   : "memory");
    }
    for (int i = vtail + tid; i < V; i += BLOCK) {
      asm volatile("global_load_async_to_lds_b32 %0, %1, off"
                   :
                   : "v"(lbase + 4u * (unsigned)i), "v"(gno + i)
                   : "memory");
    }
#endif
    async_wait_zero();
  }
  __syncthreads();

  // ---- Init: phases = noise + 0.1*logits; accumulate sums of sin/cos ----
  // Element-strided scalar pass: logits rows are only 4B-aligned, and this
  // pass is HBM-bandwidth-bound; coalesced b32 is optimal.
  float sS = 0.f, sC = 0.f;
  for (int i = tid; i < V; i += BLOCK) {
    float p = ph[i] + 0.1f * glg[i];
    ph[i] = p;
    sS += __sinf(p);
    sC += __cosf(p);
  }
  blockReduce2(sS, sC, redA, redB, bcast);  // barrier: safe to switch layout
  float S = sS, C = sC;  // row sums of sin / cos at current step

  // ---- 10 fused Kuramoto steps (chunked: ds_load/store_b128 + b128 omega) --
  // phases += DT*omega + (DT*coupling)*(sin(p)*C - cos(p)*S)
  const float Kc = (KUR_DT * 0.1f) / (float)V;  // DT * KURAMOTO_COUPLING / V
  float4* __restrict__ ph4 = reinterpret_cast<float4*>(ph);
  const float4* __restrict__ om4 = reinterpret_cast<const float4*>(omega);
  for (int t = 0; t < NSTEPS; ++t) {
    float nS = 0.f, nC = 0.f;
    for (int c = tid; c < nchunk; c += BLOCK) {
      float4 p  = ph4[c];   // ds_load_b128
      float4 om = om4[c];   // global_load_b128, 16B-aligned, L2-resident
      {
        float s = __sinf(p.x), cc = __cosf(p.x);
        p.x += KUR_DT * om.x + Kc * (s * C - cc * S);
        nS += __sinf(p.x); nC += __cosf(p.x);
      }
      {
        float s = __sinf(p.y), cc = __cosf(p.y);
        p.y += KUR_DT * om.y + Kc * (s * C - cc * S);
        nS += __sinf(p.y); nC += __cosf(p.y);
      }
      {
        float s = __sinf(p.z), cc = __cosf(p.z);
        p.z += KUR_DT * om.z + Kc * (s * C - cc * S);
        nS += __sinf(p.z); nC += __cosf(p.z);
      }
      {
        float s = __sinf(p.w), cc = __cosf(p.w);
        p.w += KUR_DT * om.w + Kc * (s * C - cc * S);
        nS += __sinf(p.w); nC += __cosf(p.w);
      }
      ph4[c] = p;           // ds_store_b128
    }
    for (int i = vtail + tid; i < V; i += BLOCK) {  // tail (1 element)
      float p = ph[i];
      float s = __sinf(p), cc = __cosf(p);
      p += KUR_DT * omega[i] + Kc * (s * C - cc * S);
      ph[i] = p;
      nS += __sinf(p);
      nC += __cosf(p);
    }
    blockReduce2(nS, nC, redA, redB, bcast);
    S = nS;
    C = nC;
  }

  // ---- coherence = cos(p - mean); softmax over the row ----
  float pSum = 0.f, zero = 0.f;
  for (int c = tid; c < nchunk; c += BLOCK) {
    float4 p = ph4[c];
    pSum += (p.x + p.y) + (p.z + p.w);
  }
  for (int i = vtail + tid; i < V; i += BLOCK) pSum += ph[i];
  blockReduce2(pSum, zero, redA, redB, bcast);
  const float mean = pSum / (float)V;

  // coherence is in [-1,1]; shift by constant 1.0 (softmax is shift-invariant)
  float eSum = 0.f, zero2 = 0.f;
  for (int c = tid; c < nchunk; c += BLOCK) {
    float4 p = ph4[c];
    p.x = __expf(__cosf(p.x - mean) - 1.0f);
    p.y = __expf(__cosf(p.y - mean) - 1.0f);
    p.z = __expf(__cosf(p.z - mean) - 1.0f);
    p.w = __expf(__cosf(p.w - mean) - 1.0f);
    eSum += (p.x + p.y) + (p.z + p.w);
    ph4[c] = p;  // reuse LDS row as exp storage
  }
  for (int i = vtail + tid; i < V; i += BLOCK) {
    float e = __expf(__cosf(ph[i] - mean) - 1.0f);
    ph[i] = e;
    eSum += e;
  }
  blockReduce2(eSum, zero2, redA, redB, bcast);  // barrier: safe to switch layout
  const float inv = 1.0f / eSum;

  // Element-strided coalesced b32 stores (output rows are only 4B-aligned).
  for (int i = tid; i < V; i += BLOCK) gout[i] = ph[i] * inv;
}

extern "C" void kernel_launch(void* const* d_in, const int* in_sizes, int n_in,
                              void* d_out, int out_size, void* d_ws, size_t ws_size,
                              hipStream_t stream) {
  const float* logits = (const float*)d_in[0];
  const float* omega  = (const float*)d_in[1];  // natural_frequencies [V]
  const float* noise  = (const float*)d_in[2];
  float* out = (float*)d_out;

  const int V    = in_sizes[1];           // 50257
  const int rows = in_sizes[0] / V;       // B*S = 1024
  hipLaunchKernelGGL(kuramoto_rowresident, dim3(rows), dim3(BLOCK), 0, stream,
                     logits, omega, noise, out, V);
}